// Correlation_13314398618020
// MI455X (gfx1250) — compile-verified
//
#include <hip/hip_runtime.h>

typedef float v2f __attribute__((ext_vector_type(2)));
typedef float v8f __attribute__((ext_vector_type(8)));

namespace {
constexpr int kB  = 8;
constexpr int kC  = 256;
constexpr int kH  = 96;
constexpr int kW  = 160;
constexpr int kNJ = 21;                 // displacements per axis (stride 2, max 20)
constexpr int kHW = kH * kW;            // channel-plane stride (15360)
constexpr int kNT = 4;                  // N-tiles covering band [x0-20, x0+43]
constexpr int kLS  = 208;               // LDS columns per channel row: u2 in [-20,188)
constexpr int kPR  = 2 * kLS;           // floats per K-pair row (416): [c even, c odd] interleaved
constexpr int kPR4 = kPR / 4;           // 104 float4 per pair row
constexpr int kBlocks = kB * kH * kNJ;  // 16128 blocks, one per (b, y, j)
}

// One block per (b, y, j); wave w computes x-tile w (banded-Gram 16x16 WMMA over K=C).
// Zero-padded x2 row staged in LDS, K-pair interleaved so each WMMA B fragment
// is a single aligned ds_load_b64 (no register shuffles).
__global__ __launch_bounds__(320, 1) void corr441_wmma_f32(
    const float* __restrict__ x1, const float* __restrict__ x2,
    float* __restrict__ out)
{
  __shared__ float sh[kC * kLS];        // 208 KB (CDNA5: 320 KB/WGP)

  const int tid  = threadIdx.x;
  const int lane = tid & 31;            // wave32
  const int half = lane >> 4;
  const int l    = lane & 15;
  const int xt   = tid >> 5;            // wave id == x-tile (0..9)

  int bi = blockIdx.x;
  const int j = bi % kNJ; bi /= kNJ;
  const int y = bi % kH;  bi /= kH;
  const int b = bi;

  const int y2 = y + 2 * j - 20;                           // real x2 row
  const int outRow = ((b * 441 + j * kNJ) * kH + y) * kW;  // + i*kHW + x

  if ((unsigned)y2 >= (unsigned)kH) {
    // whole (b,y,j) slab is padding -> zeros (block-uniform branch, no barrier used)
    for (int i = tid; i < kNJ * kW; i += 320)
      out[outRow + (i / kW) * kHW + (i % kW)] = 0.0f;
    return;
  }

  // ---- Stage zero-padded, K-pair-interleaved x2 row into LDS ----
  // pair-row p, column u (LDS col = u2+20): floats [p*416 + col*2 + 0] = ch 2p,
  //                                               [p*416 + col*2 + 1] = ch 2p+1
  float4* sh4 = (float4*)sh;
  const float4* src4 = (const float4*)x2;
  const int rowBase4 = ((b * kC) * kH + y2) * (kW / 4);    // + c*(kHW/4)
  constexpr int kQ = kW / 4;                               // 40 quads per channel row
  for (int i = tid; i < (kC / 2) * kQ; i += 320) {
    const int p = i / kQ;             // channel pair 0..127
    const int q = i - p * kQ;         // quad (cols 4q..4q+3)
    const float4 va = src4[rowBase4 + (2 * p)     * (kHW / 4) + q];
    const float4 vb = src4[rowBase4 + (2 * p + 1) * (kHW / 4) + q];
    const float4 lo = {va.x, vb.x, va.y, vb.y};
    const float4 hi = {va.z, vb.z, va.w, vb.w};
    sh4[p * kPR4 + 10 + 2 * q    ] = lo;                   // cols 20+4q, 20+4q+1
    sh4[p * kPR4 + 10 + 2 * q + 1] = hi;                   // cols 20+4q+2, +3
  }
  const float4 z4 = {0.f, 0.f, 0.f, 0.f};
  for (int i = tid; i < (kC / 2) * 24; i += 320) {         // cols [0,20) & [180,208)
    const int p = i / 24;
    const int q = i - p * 24;
    const int c4 = (q < 10) ? q : (90 + (q - 10));
    sh4[p * kPR4 + c4] = z4;
  }
  __syncthreads();

  // ---- Banded-Gram accumulation ----
  const int x0 = xt << 4;
  const int u2base = x0 - 20 + l;       // this lane's N coordinate, tile 0

  v8f acc0 = {}, acc1 = {}, acc2 = {}, acc3 = {};

  // A (16x4 f32): lanes 0-15 supply K=0,1; lanes 16-31 supply K=2,3 (ISA 7.12.2)
  const int kb = half << 1;
  const float* ap = x1 + ((b * kC + kb) * kH + y) * kW + x0 + l;
  // B (4x16 f32): fragment {K=kb, K=kb+1} at col x0+l+16t = one ds_load_b64.
  // chunk c0 -> pair-row c0/4 + half; per-chunk advance = 1 pair row = 208 v2f.
  const v2f* sp = (const v2f*)(sh + half * kPR) + (x0 + l);

#pragma unroll 4
  for (int c0 = 0; c0 < kC; c0 += 4) {
    v2f a  = { ap[0], ap[kHW] };
    v2f b0 = sp[0];                     // tile 0 (col +0)
    v2f b1 = sp[16];                    // tile 1 (col +16)
    v2f b2 = sp[32];                    // tile 2 (col +32)
    v2f b3 = sp[48];                    // tile 3 (col +48)

    acc0 = __builtin_amdgcn_wmma_f32_16x16x4_f32(false, a, false, b0, (short)0, acc0, false, false);
    acc1 = __builtin_amdgcn_wmma_f32_16x16x4_f32(false, a, false, b1, (short)0, acc1, false, false);
    acc2 = __builtin_amdgcn_wmma_f32_16x16x4_f32(false, a, false, b2, (short)0, acc2, false, false);
    acc3 = __builtin_amdgcn_wmma_f32_16x16x4_f32(false, a, false, b3, (short)0, acc3, false, false);

    ap += 4 * kHW;
    sp += kLS;                          // 1 pair-row = 416 floats = 208 v2f
  }

  // ---- Scatter the valid band: d = u2 - x + 20 must be even and in [0, 40] ----
  const float invC = 1.0f / (float)kC;
  v8f accs[kNT] = {acc0, acc1, acc2, acc3};
#pragma unroll
  for (int t = 0; t < kNT; ++t) {
    const int u2 = u2base + 16 * t;       // lane's N coordinate (C/D: N = lane%16)
#pragma unroll
    for (int r = 0; r < 8; ++r) {
      const int x = x0 + r + (half << 3); // C/D: M = r + 8*(lane/16)
      const int d = u2 - x + 20;
      if (((d & 1) == 0) && ((unsigned)d <= 40u)) {
        out[outRow + (d >> 1) * kHW + x] = accs[t][r] * invC;
      }
    }
  }
}

extern "C" void kernel_launch(void* const* d_in, const int* in_sizes, int n_in,
                              void* d_out, int out_size, void* d_ws, size_t ws_size,
                              hipStream_t stream) {
  (void)in_sizes; (void)n_in; (void)d_ws; (void)ws_size; (void)out_size;
  const float* x1 = (const float*)d_in[0];
  const float* x2 = (const float*)d_in[1];
  float* out = (float*)d_out;

  dim3 block(320);                       // 10 waves: one per x-tile (wave32)
  dim3 grid(kBlocks);                    // 16128: one block per (b, y, j)
  hipLaunchKernelGGL(corr441_wmma_f32, grid, block, 0, stream, x1, x2, out);
}